// AttnDecoderRNN_4011499455146
// MI455X (gfx1250) — compile-verified
//
#include <hip/hip_runtime.h>
#include <hip/hip_bf16.h>
#include <cmath>

typedef __attribute__((ext_vector_type(2))) float v2f;
typedef __attribute__((ext_vector_type(8))) float v8f;
typedef __attribute__((ext_vector_type(4))) int v4i;

#define HDIM 512
#define BDIM 64
#define SDIM 2048
#define VDIM 32000

#if defined(__has_builtin)
#if __has_builtin(__builtin_amdgcn_global_load_async_to_lds_b128) && \
    __has_builtin(__builtin_amdgcn_s_wait_asynccnt)
#define USE_ASYNC_LDS 1
typedef __attribute__((address_space(1))) v4i* gptr_v4i;
typedef __attribute__((address_space(3))) v4i* lptr_v4i;
#endif
#endif

// ---------------------------------------------------------------------------
// Generic fp32 WMMA GEMM:  C[m,n] = sum_k A[m,k] * B[n,k] + bias[n]
// A: (M, lda) row-major, B: (N, ldb) row-major ("weight" layout), C: (M, ldc)
// grid = (M/16, N/64), block = 128 threads (4 waves), wave w -> N-tile w.
// V_WMMA_F32_16X16X4_F32 fragment layout (wave32):
//   A (16x4): lane<16 -> row=lane,   k = k0+{0,1};  lane>=16 -> row=lane-16, k = k0+{2,3}
//   B (4x16): lane<16 -> col=lane,   k = k0+{0,1};  lane>=16 -> col=lane-16, k = k0+{2,3}
//   D (16x16): elem i of lane L -> m = (L>>4)*8 + i, n = L&15
// ---------------------------------------------------------------------------
__global__ __launch_bounds__(128) void wmma_gemm_bias(
    const float* __restrict__ A, int lda,
    const float* __restrict__ B, int ldb,
    const float* __restrict__ bias,
    float* __restrict__ C, int ldc,
    int K)
{
  const int lane = threadIdx.x & 31;
  const int wave = threadIdx.x >> 5;
  const int m0   = blockIdx.x * 16;
  const int n0   = blockIdx.y * 64 + wave * 16;
  const int row  = lane & 15;
  const int koff = (lane >> 4) * 2;

  const float* ap = A + (size_t)(m0 + row) * lda + koff;
  const float* bp = B + (size_t)(n0 + row) * ldb + koff;

  v8f acc = {};
  for (int k = 0; k < K; k += 4) {
    v2f a = *(const v2f*)(ap + k);
    v2f b = *(const v2f*)(bp + k);
    acc = __builtin_amdgcn_wmma_f32_16x16x4_f32(false, a, false, b,
                                                (short)0, acc, false, false);
  }

  const float bz = bias ? bias[n0 + row] : 0.0f;
  const int mbase = m0 + ((lane >> 4) << 3);
  float* cp = C + (size_t)mbase * ldc + n0 + row;
#pragma unroll
  for (int i = 0; i < 8; ++i)
    cp[(size_t)i * ldc] = acc[i] + bz;
}

// ---------------------------------------------------------------------------
// Fused energy + scores kernel (the 68.7 GFLOP pole).
// One block per (s, b0) with b0 in {0, 32}: 32 rows (2 M-tiles) per block.
//   scores[b0+m, s] = sum_k tanh( q[b0+m,k]
//                                 + sum_h enc[s,b0+m,h]*We[k,h] ) * v[k]
// A-tile (32 x 512) staged in LDS via async-load-to-LDS (ASYNCcnt path),
// row pad 516 floats to spread 16x4 fragment reads across banks.
// 8 waves x 4 N-tiles x 2 M-tiles: per k-step = 2 LDS A-loads + 4 global
// B-loads + 8 WMMA (0.75 loads/WMMA); We L2 traffic halved vs 16-row tile.
// ---------------------------------------------------------------------------
#define APAD 516

__global__ __launch_bounds__(256) void attn_scores_kernel(
    const float* __restrict__ enc,    // (S, B, H)
    const float* __restrict__ attn_W, // (H, 2H); We = cols [H, 2H)
    const float* __restrict__ q,      // (B, H) = hid@Wa^T + attn_b
    const float* __restrict__ vvec,   // (H)
    float* __restrict__ scores)       // (B, S)
{
  __shared__ float As[32 * APAD];
  __shared__ float red[8][32];

  const int bid = blockIdx.x;
  const int s   = bid >> 1;
  const int b0  = (bid & 1) << 5;
  const int tid = threadIdx.x;

  // Stage A-tile: 32 rows x 512 floats; 8 threads per row, 16 x b128 each.
  {
    const int r = tid >> 3;
    const int c = (tid & 7) * 64;
    const float* g = enc + ((size_t)s * BDIM + b0 + r) * HDIM + c;
    float* l = As + r * APAD + c;
#if USE_ASYNC_LDS
#pragma unroll
    for (int i = 0; i < 16; ++i)
      __builtin_amdgcn_global_load_async_to_lds_b128(
          (gptr_v4i)(g + i * 4), (lptr_v4i)(l + i * 4), 0, 0);
    __builtin_amdgcn_s_wait_asynccnt(0);
#else
#pragma unroll
    for (int i = 0; i < 16; ++i)
      *(float4*)(l + i * 4) = *(const float4*)(g + i * 4);
#endif
  }
  __syncthreads();

  const int lane = tid & 31;
  const int wave = tid >> 5;
  const int row  = lane & 15;
  const int half = lane >> 4;       // 0 or 1
  const int koff = half * 2;

  const float* lap0 = As + row * APAD + koff;          // M-tile 0 (rows 0..15)
  const float* lap1 = As + (16 + row) * APAD + koff;   // M-tile 1 (rows 16..31)

  const float* bp0 = attn_W + (size_t)((wave * 4 + 0) * 16 + row) * (2 * HDIM) + HDIM + koff;
  const float* bp1 = attn_W + (size_t)((wave * 4 + 1) * 16 + row) * (2 * HDIM) + HDIM + koff;
  const float* bp2 = attn_W + (size_t)((wave * 4 + 2) * 16 + row) * (2 * HDIM) + HDIM + koff;
  const float* bp3 = attn_W + (size_t)((wave * 4 + 3) * 16 + row) * (2 * HDIM) + HDIM + koff;

  v8f acc[2][4];
#pragma unroll
  for (int mt = 0; mt < 2; ++mt)
#pragma unroll
    for (int nt = 0; nt < 4; ++nt)
      acc[mt][nt] = (v8f){};

  for (int k = 0; k < HDIM; k += 4) {
    v2f a0 = *(const v2f*)(lap0 + k);
    v2f a1 = *(const v2f*)(lap1 + k);
    v2f b0v = *(const v2f*)(bp0 + k);
    v2f b1v = *(const v2f*)(bp1 + k);
    v2f b2v = *(const v2f*)(bp2 + k);
    v2f b3v = *(const v2f*)(bp3 + k);
    acc[0][0] = __builtin_amdgcn_wmma_f32_16x16x4_f32(false, a0, false, b0v, (short)0, acc[0][0], false, false);
    acc[1][0] = __builtin_amdgcn_wmma_f32_16x16x4_f32(false, a1, false, b0v, (short)0, acc[1][0], false, false);
    acc[0][1] = __builtin_amdgcn_wmma_f32_16x16x4_f32(false, a0, false, b1v, (short)0, acc[0][1], false, false);
    acc[1][1] = __builtin_amdgcn_wmma_f32_16x16x4_f32(false, a1, false, b1v, (short)0, acc[1][1], false, false);
    acc[0][2] = __builtin_amdgcn_wmma_f32_16x16x4_f32(false, a0, false, b2v, (short)0, acc[0][2], false, false);
    acc[1][2] = __builtin_amdgcn_wmma_f32_16x16x4_f32(false, a1, false, b2v, (short)0, acc[1][2], false, false);
    acc[0][3] = __builtin_amdgcn_wmma_f32_16x16x4_f32(false, a0, false, b3v, (short)0, acc[0][3], false, false);
    acc[1][3] = __builtin_amdgcn_wmma_f32_16x16x4_f32(false, a1, false, b3v, (short)0, acc[1][3], false, false);
  }

  // Epilogue: tanh(acc + q) * v[k], accumulate per-row partial scores.
  float part[2][8];
#pragma unroll
  for (int mt = 0; mt < 2; ++mt)
#pragma unroll
    for (int i = 0; i < 8; ++i) part[mt][i] = 0.0f;

#pragma unroll
  for (int nt = 0; nt < 4; ++nt) {
    const int n0 = (wave * 4 + nt) * 16;
    const float vk = vvec[n0 + row];
#pragma unroll
    for (int mt = 0; mt < 2; ++mt) {
      const float* qp = q + (size_t)(b0 + mt * 16 + half * 8) * HDIM + n0 + row;
#pragma unroll
      for (int i = 0; i < 8; ++i)
        part[mt][i] += tanhf(acc[mt][nt][i] + qp[(size_t)i * HDIM]) * vk;
    }
  }

  // Reduce over n (16 lanes within each half of the wave).
#pragma unroll
  for (int off = 1; off < 16; off <<= 1)
#pragma unroll
    for (int mt = 0; mt < 2; ++mt)
#pragma unroll
      for (int i = 0; i < 8; ++i)
        part[mt][i] += __shfl_xor(part[mt][i], off, 32);

  if (row == 0) {
#pragma unroll
    for (int mt = 0; mt < 2; ++mt)
#pragma unroll
      for (int i = 0; i < 8; ++i)
        red[wave][mt * 16 + half * 8 + i] = part[mt][i];
  }
  __syncthreads();

  if (tid < 32) {
    float acc2 = 0.0f;
#pragma unroll
    for (int w = 0; w < 8; ++w) acc2 += red[w][tid];
    scores[(size_t)(b0 + tid) * SDIM + s] = acc2;
  }
}

// ---------------------------------------------------------------------------
// Row softmax over S=2048 per batch row. scores (B,S) -> attnw (B,S).
// ---------------------------------------------------------------------------
__global__ __launch_bounds__(256) void softmax_kernel(
    const float* __restrict__ scores, float* __restrict__ attnw)
{
  __shared__ float buf[SDIM];
  __shared__ float rbuf[8];
  const int b = blockIdx.x, tid = threadIdx.x;
  const float* srow = scores + (size_t)b * SDIM;

  float lmax = -3.4e38f;
  for (int s = tid; s < SDIM; s += 256) {
    float x = srow[s]; buf[s] = x; lmax = fmaxf(lmax, x);
  }
  for (int off = 16; off; off >>= 1) lmax = fmaxf(lmax, __shfl_xor(lmax, off, 32));
  if ((tid & 31) == 0) rbuf[tid >> 5] = lmax;
  __syncthreads();
  float bmax = rbuf[0];
#pragma unroll
  for (int i = 1; i < 8; ++i) bmax = fmaxf(bmax, rbuf[i]);

  float lsum = 0.0f;
  for (int s = tid; s < SDIM; s += 256) {
    float e = expf(buf[s] - bmax); buf[s] = e; lsum += e;
  }
  for (int off = 16; off; off >>= 1) lsum += __shfl_xor(lsum, off, 32);
  __syncthreads();
  if ((tid & 31) == 0) rbuf[tid >> 5] = lsum;
  __syncthreads();
  float bsum = 0.0f;
#pragma unroll
  for (int i = 0; i < 8; ++i) bsum += rbuf[i];
  const float inv = 1.0f / bsum;

  for (int s = tid; s < SDIM; s += 256)
    attnw[(size_t)b * SDIM + s] = buf[s] * inv;
}

// ---------------------------------------------------------------------------
// Context partials: grid (B, 32); block c accumulates 64 s-values.
// part[c][b][h] = sum_{s in chunk} attnw[b,s] * enc[s,b,h]
// ---------------------------------------------------------------------------
__global__ __launch_bounds__(512) void context_part_kernel(
    const float* __restrict__ attnw, const float* __restrict__ enc,
    float* __restrict__ part)
{
  const int b = blockIdx.x, c = blockIdx.y, h = threadIdx.x;
  const float* ep = enc + ((size_t)(c * 64) * BDIM + b) * HDIM + h;
  const float* wp = attnw + (size_t)b * SDIM + c * 64;
  float acc = 0.0f;
#pragma unroll 4
  for (int s = 0; s < 64; ++s)
    acc += wp[s] * ep[(size_t)s * (BDIM * HDIM)];
  part[((size_t)c * BDIM + b) * HDIM + h] = acc;
}

// Reduce 32 partials; assemble x = [emb[word]; ctx], y[:,512:] = ctx.
__global__ __launch_bounds__(512) void context_reduce_kernel(
    const float* __restrict__ part, const float* __restrict__ emb,
    const int* __restrict__ word, float* __restrict__ x, float* __restrict__ y)
{
  const int b = blockIdx.x, h = threadIdx.x;
  float acc = 0.0f;
#pragma unroll
  for (int c = 0; c < 32; ++c)
    acc += part[((size_t)c * BDIM + b) * HDIM + h];
  x[(size_t)b * 1024 + HDIM + h] = acc;
  y[(size_t)b * 1024 + HDIM + h] = acc;
  x[(size_t)b * 1024 + h] = emb[(size_t)word[b] * HDIM + h];
}

// ---------------------------------------------------------------------------
// GRU gates: h_new -> d_out tail and y[:, :512].
// ---------------------------------------------------------------------------
__global__ __launch_bounds__(256) void gru_gate_kernel(
    const float* __restrict__ gi, const float* __restrict__ gh,
    const float* __restrict__ hid, float* __restrict__ hnew_out,
    float* __restrict__ y)
{
  const int idx = blockIdx.x * 256 + threadIdx.x;   // B*H threads
  const int b = idx >> 9, j = idx & 511;
  const float* gib = gi + (size_t)b * 1536;
  const float* ghb = gh + (size_t)b * 1536;
  float r = 1.0f / (1.0f + expf(-(gib[j] + ghb[j])));
  float z = 1.0f / (1.0f + expf(-(gib[512 + j] + ghb[512 + j])));
  float n = tanhf(gib[1024 + j] + r * ghb[1024 + j]);
  float h = (1.0f - z) * n + z * hid[(size_t)b * HDIM + j];
  hnew_out[(size_t)b * HDIM + j] = h;
  y[(size_t)b * 1024 + j] = h;
}

// ---------------------------------------------------------------------------
// In-place row log-softmax over V=32000 (on d_out).
// ---------------------------------------------------------------------------
__global__ __launch_bounds__(1024) void log_softmax_kernel(float* __restrict__ out)
{
  __shared__ float rbuf[32];
  const int b = blockIdx.x, tid = threadIdx.x;
  float* row = out + (size_t)b * VDIM;

  float lmax = -3.4e38f;
  for (int s = tid; s < VDIM; s += 1024) lmax = fmaxf(lmax, row[s]);
  for (int off = 16; off; off >>= 1) lmax = fmaxf(lmax, __shfl_xor(lmax, off, 32));
  if ((tid & 31) == 0) rbuf[tid >> 5] = lmax;
  __syncthreads();
  float bmax = rbuf[0];
#pragma unroll
  for (int i = 1; i < 32; ++i) bmax = fmaxf(bmax, rbuf[i]);

  float lsum = 0.0f;
  for (int s = tid; s < VDIM; s += 1024) lsum += expf(row[s] - bmax);
  for (int off = 16; off; off >>= 1) lsum += __shfl_xor(lsum, off, 32);
  __syncthreads();
  if ((tid & 31) == 0) rbuf[tid >> 5] = lsum;
  __syncthreads();
  float bsum = 0.0f;
#pragma unroll
  for (int i = 0; i < 32; ++i) bsum += rbuf[i];
  const float lse = bmax + logf(bsum);

  for (int s = tid; s < VDIM; s += 1024) row[s] -= lse;
}

// ---------------------------------------------------------------------------
extern "C" void kernel_launch(void* const* d_in, const int* in_sizes, int n_in,
                              void* d_out, int out_size, void* d_ws, size_t ws_size,
                              hipStream_t stream)
{
  (void)in_sizes; (void)n_in; (void)out_size; (void)ws_size;

  const int*   word   = (const int*)  d_in[0];
  const float* hid    = (const float*)d_in[1];   // (1,B,H) == (B,H)
  const float* enc    = (const float*)d_in[2];   // (S,B,H)
  const float* emb    = (const float*)d_in[3];   // (V,H)
  const float* attn_W = (const float*)d_in[4];   // (H,2H)
  const float* attn_b = (const float*)d_in[5];   // (H)
  const float* vvec   = (const float*)d_in[6];   // (H)
  const float* W_ih   = (const float*)d_in[7];   // (3H, H+E=1024)
  const float* W_hh   = (const float*)d_in[8];   // (3H, H)
  const float* b_ih   = (const float*)d_in[9];
  const float* b_hh   = (const float*)d_in[10];
  const float* out_W  = (const float*)d_in[11];  // (V, 2H)
  const float* out_b  = (const float*)d_in[12];

  float* out_logits = (float*)d_out;                       // (B, V)
  float* out_hnew   = (float*)d_out + (size_t)BDIM * VDIM; // (B, H)

  // Workspace layout (floats).
  float* ws      = (float*)d_ws;
  float* q       = ws;                       // B*H        = 32768
  float* scores  = q      + BDIM * HDIM;     // B*S        = 131072
  float* attnw   = scores + BDIM * SDIM;     // B*S        = 131072
  float* x       = attnw  + BDIM * SDIM;     // B*1024     = 65536
  float* y       = x      + BDIM * 1024;     // B*1024     = 65536
  float* gi      = y      + BDIM * 1024;     // B*1536     = 98304
  float* gh      = gi     + BDIM * 1536;     // B*1536     = 98304
  float* ctxpart = gh     + BDIM * 1536;     // 32*B*H     = 1048576

  // 1) q = hid @ Wa^T + attn_b   (Wa = attn_W[:, :H], ldb = 2H)
  wmma_gemm_bias<<<dim3(BDIM / 16, HDIM / 64), 128, 0, stream>>>(
      hid, HDIM, attn_W, 2 * HDIM, attn_b, q, HDIM, HDIM);

  // 2) fused energy->scores (dominant GEMM, 68.7 GFLOP); 2 blocks per s
  attn_scores_kernel<<<SDIM * (BDIM / 32), 256, 0, stream>>>(
      enc, attn_W, q, vvec, scores);

  // 3) softmax over S
  softmax_kernel<<<BDIM, 256, 0, stream>>>(scores, attnw);

  // 4) context = attn . enc (2-stage for HBM parallelism), build x / y-tail
  context_part_kernel<<<dim3(BDIM, 32), 512, 0, stream>>>(attnw, enc, ctxpart);
  context_reduce_kernel<<<BDIM, 512, 0, stream>>>(ctxpart, emb, word, x, y);

  // 5) GRU gate pre-activations
  wmma_gemm_bias<<<dim3(BDIM / 16, 1536 / 64), 128, 0, stream>>>(
      x, 1024, W_ih, 1024, b_ih, gi, 1536, 1024);
  wmma_gemm_bias<<<dim3(BDIM / 16, 1536 / 64), 128, 0, stream>>>(
      hid, HDIM, W_hh, HDIM, b_hh, gh, 1536, HDIM);

  // 6) gates -> h_new (d_out tail) and y[:, :512]
  gru_gate_kernel<<<(BDIM * HDIM) / 256, 256, 0, stream>>>(
      gi, gh, hid, out_hnew, y);

  // 7) logits = y @ out_W^T + out_b  -> d_out, then in-place log-softmax
  wmma_gemm_bias<<<dim3(BDIM / 16, VDIM / 64), 128, 0, stream>>>(
      y, 1024, out_W, 1024, out_b, out_logits, VDIM, 1024);
  log_softmax_kernel<<<BDIM, 1024, 0, stream>>>(out_logits);
}